// MeanMaxAggregator_85220741087351
// MI455X (gfx1250) — compile-verified
//
#include <hip/hip_runtime.h>
#include <stdint.h>

#define NSEG   64
#define DCOLS  256
#define TROWS  32
#define TILE_F (TROWS * DCOLS)   /* 8192 floats */
#define TILE_B (TILE_F * 4)      /* 32768 bytes */

typedef float v2f __attribute__((ext_vector_type(2)));
typedef float v8f __attribute__((ext_vector_type(8)));

__device__ __forceinline__ unsigned fmap(float f) {
  unsigned b = __float_as_uint(f);
  return (b & 0x80000000u) ? ~b : (b | 0x80000000u);
}
__device__ __forceinline__ float funmap(unsigned u) {
  unsigned b = (u & 0x80000000u) ? (u & 0x7fffffffu) : ~u;
  return __uint_as_float(b);
}

// ---------------- init: zero accumulators + binary-search segment boundaries --
__global__ void __launch_bounds__(256)
k_init(const long long* __restrict__ idx, int N, int* __restrict__ bnd,
       float* __restrict__ sums, unsigned* __restrict__ maxu) {
  int g = blockIdx.x * 256 + threadIdx.x;
  if (g < NSEG * DCOLS) { sums[g] = 0.0f; maxu[g] = 0u; }
  if (g <= NSEG) {
    long long v = (long long)g;
    int lo = 0, hi = N;
    while (lo < hi) { int mid = (lo + hi) >> 1; if (idx[mid] < v) lo = mid + 1; else hi = mid; }
    bnd[g] = lo;
  }
}

// ---------------- main streaming kernel --------------------------------------
__global__ void __launch_bounds__(256)
k_main(const float* __restrict__ feat, int N, int rowsPerBlk,
       const int* __restrict__ bndG, float* __restrict__ sums,
       unsigned* __restrict__ maxu) {
  __shared__ __align__(16) float tile[2 * TILE_F];
  __shared__ int bnd[NSEG + 2];

  const int tid  = threadIdx.x;
  const int lane = tid & 31;
  const int wave = tid >> 5;
  const bool hi  = lane >= 16;
  const int m    = lane & 15;

  const int r0 = blockIdx.x * rowsPerBlk;
  if (r0 >= N) return;
  const int rend = min(r0 + rowsPerBlk, N);
  const int T    = (rend - r0 + TROWS - 1) / TROWS;

  if (tid <= NSEG) bnd[tid] = bndG[tid];
  __syncthreads();

  const unsigned ldsBase = (unsigned)(unsigned long long)(&tile[0]);
  const unsigned long long maxByte = (unsigned long long)N * (DCOLS * 4ull);

  // issue tile 0 (async global -> LDS, 16B per lane per op, 8 ops/thread)
  {
    unsigned gb = (unsigned)r0 * (DCOLS * 4u);
#pragma unroll
    for (int i = 0; i < 8; ++i) {
      unsigned ix   = (unsigned)(i * 256 + tid);
      unsigned goff = gb + ix * 16u;
      if ((unsigned long long)goff + 16ull > maxByte) goff = 0u;
      unsigned loff = ldsBase + ix * 16u;
      asm volatile("global_load_async_to_lds_b128 %0, %1, %2"
                   :: "v"(loff), "v"(goff), "s"(feat) : "memory");
    }
  }

  const int c0 = wave * 32 + m;   // column of first 16-wide WMMA tile
  const int c1 = c0 + 16;        // column of second tile

  v8f acc0 = {0,0,0,0,0,0,0,0};
  v8f acc1 = {0,0,0,0,0,0,0,0};

  int cur = 0;
  while (cur < NSEG - 1 && r0 >= bnd[cur + 1]) cur++;
  int seg_base = cur;   // base of 16-slot WMMA window
  int maxslot  = 0;

  int   mseg  = -1;
  float vmax0 = -3.0e38f, vmax1 = -3.0e38f;

  auto flush_sums = [&]() {
    const int mofs = hi ? 8 : 0;
    const int used = maxslot + 1;
#pragma unroll
    for (int v = 0; v < 8; ++v) {
      int M = v + mofs;
      if (M < used) {
        float x0 = acc0[v], x1 = acc1[v];
        if (x0 != 0.0f) atomicAdd(&sums[(seg_base + M) * DCOLS + c0], x0);
        if (x1 != 0.0f) atomicAdd(&sums[(seg_base + M) * DCOLS + c1], x1);
      }
      acc0[v] = 0.0f; acc1[v] = 0.0f;
    }
  };
  auto flush_max = [&]() {
    if (mseg >= 0) {
      atomicMax(&maxu[mseg * DCOLS + c0], fmap(vmax0));
      atomicMax(&maxu[mseg * DCOLS + c1], fmap(vmax1));
    }
    vmax0 = -3.0e38f; vmax1 = -3.0e38f;
  };
  auto segOf = [&](int row) {
    while (cur < NSEG - 1 && row >= bnd[cur + 1]) cur++;
    return cur;
  };

  for (int t = 0; t < T; ++t) {
    // prefetch next tile into the other buffer, then wait for current tile
    if (t + 1 < T) {
      unsigned gb    = (unsigned)(r0 + (t + 1) * TROWS) * (DCOLS * 4u);
      unsigned lbase = ldsBase + (unsigned)((t + 1) & 1) * TILE_B;
#pragma unroll
      for (int i = 0; i < 8; ++i) {
        unsigned ix   = (unsigned)(i * 256 + tid);
        unsigned goff = gb + ix * 16u;
        if ((unsigned long long)goff + 16ull > maxByte) goff = 0u;
        unsigned loff = lbase + ix * 16u;
        asm volatile("global_load_async_to_lds_b128 %0, %1, %2"
                     :: "v"(loff), "v"(goff), "s"(feat) : "memory");
      }
      asm volatile("s_wait_asynccnt 0x8" ::: "memory");  // this wave's tile-t ops done
    } else {
      asm volatile("s_wait_asynccnt 0x0" ::: "memory");
    }
    __syncthreads();  // all waves' contributions to tile t visible

    const float* tb = &tile[(t & 1) * TILE_F];
    const int rsBase = r0 + t * TROWS;

#pragma unroll
    for (int rt = 0; rt < 8; ++rt) {
      int r = rsBase + rt * 4;
      bool v0 = (r + 0) < rend, v1 = (r + 1) < rend,
           v2 = (r + 2) < rend, v3 = (r + 3) < rend;
      if (!v0) break;  // uniform

      int s0 = segOf(r);
      int s1 = v1 ? segOf(r + 1) : s0;
      int s2 = v2 ? segOf(r + 2) : s1;
      int s3 = v3 ? segOf(r + 3) : s2;

      // B operand: 4x16 f32 -> VGPR0={K0|K2}, VGPR1={K1|K3} across lane halves
      int kk = hi ? 2 : 0;
      int rl = rt * 4 + kk;
      float bx0 = tb[(rl)     * DCOLS + c0];
      float by0 = tb[(rl + 1) * DCOLS + c0];
      float bx1 = tb[(rl)     * DCOLS + c1];
      float by1 = tb[(rl + 1) * DCOLS + c1];

      // -------- segment sum via one-hot WMMA f32 16x16x4 --------
      if (s3 - seg_base <= 15) {  // fast path (virtually always)
        int  ma = (hi ? s2 : s0) - seg_base;
        int  mb = (hi ? s3 : s1) - seg_base;
        bool va = hi ? v2 : v0, vb = hi ? v3 : v1;
        v2f a;  a.x = (va && ma == m) ? 1.0f : 0.0f;
                a.y = (vb && mb == m) ? 1.0f : 0.0f;
        v2f b0; b0.x = bx0; b0.y = by0;
        v2f b1; b1.x = bx1; b1.y = by1;
        acc0 = __builtin_amdgcn_wmma_f32_16x16x4_f32(false, a, false, b0, (short)0, acc0, false, false);
        acc1 = __builtin_amdgcn_wmma_f32_16x16x4_f32(false, a, false, b1, (short)0, acc1, false, false);
        int lastv = v3 ? s3 : (v2 ? s2 : (v1 ? s1 : s0));
        maxslot = max(maxslot, lastv - seg_base);
      } else {  // rare: segment window overflow -> row at a time with rebase
#define PROC1(K, SK, VK)                                                       \
        if (VK) {                                                              \
          if ((SK) - seg_base > 15) { flush_sums(); seg_base = (SK); maxslot = 0; } \
          int  slot = (SK) - seg_base;                                         \
          bool mine = hi ? ((K) >= 2) : ((K) < 2);                             \
          bool low  = ((K) & 1) == 0;                                          \
          float one = (mine && slot == m) ? 1.0f : 0.0f;                       \
          v2f a;  a.x = low ? one : 0.0f; a.y = low ? 0.0f : one;              \
          v2f b0; b0.x = bx0; b0.y = by0;                                      \
          v2f b1; b1.x = bx1; b1.y = by1;                                      \
          acc0 = __builtin_amdgcn_wmma_f32_16x16x4_f32(false, a, false, b0, (short)0, acc0, false, false); \
          acc1 = __builtin_amdgcn_wmma_f32_16x16x4_f32(false, a, false, b1, (short)0, acc1, false, false); \
          maxslot = max(maxslot, slot);                                        \
        }
        PROC1(0, s0, v0) PROC1(1, s1, v1) PROC1(2, s2, v2) PROC1(3, s3, v3)
#undef PROC1
      }

      // -------- segment max on VALU (co-executes with WMMA) --------
      {
        int sa = hi ? s2 : s0; bool va = hi ? v2 : v0;
        int sb = hi ? s3 : s1; bool vb = hi ? v3 : v1;
        if (va) {
          if (sa != mseg) { flush_max(); mseg = sa; }
          vmax0 = fmaxf(vmax0, bx0); vmax1 = fmaxf(vmax1, bx1);
        }
        if (vb) {
          if (sb != mseg) { flush_max(); mseg = sb; }
          vmax0 = fmaxf(vmax0, by0); vmax1 = fmaxf(vmax1, by1);
        }
      }
    }
    __syncthreads();  // done reading this buffer before it is re-filled
  }

  flush_sums();
  flush_max();
}

// ---------------- finalize: mean + unmap max, mask empty segments ------------
__global__ void __launch_bounds__(256)
k_final(const int* __restrict__ bnd, const float* __restrict__ sums,
        const unsigned* __restrict__ maxu, float* __restrict__ out) {
  int s = blockIdx.x, c = threadIdx.x;
  int cnt = bnd[s + 1] - bnd[s];
  float mean = 0.0f, mx = 0.0f;
  if (cnt > 0) {
    mean = sums[s * DCOLS + c] / (float)cnt;
    mx   = funmap(maxu[s * DCOLS + c]);
  }
  out[s * (2 * DCOLS) + c]         = mean;
  out[s * (2 * DCOLS) + DCOLS + c] = mx;
}

extern "C" void kernel_launch(void* const* d_in, const int* in_sizes, int n_in,
                              void* d_out, int out_size, void* d_ws, size_t ws_size,
                              hipStream_t stream) {
  const float*     feat = (const float*)d_in[0];
  const long long* bidx = (const long long*)d_in[1];  // int64 per reference
  int N = in_sizes[1];

  char*     ws   = (char*)d_ws;
  int*      bnd  = (int*)ws;                                   // 65 ints
  float*    sums = (float*)(ws + 512);                         // 64*256 f32
  unsigned* maxu = (unsigned*)(ws + 512 + NSEG * DCOLS * 4);   // 64*256 u32
  float*    out  = (float*)d_out;

  k_init<<<64, 256, 0, stream>>>(bidx, N, bnd, sums, maxu);

  const int targetBlocks = 1024;
  int rpb = (N + targetBlocks - 1) / targetBlocks;
  rpb = ((rpb + TROWS - 1) / TROWS) * TROWS;
  if (rpb < TROWS) rpb = TROWS;
  int nblk = (N + rpb - 1) / rpb;

  k_main<<<nblk, 256, 0, stream>>>(feat, N, rpb, bnd, sums, maxu);
  k_final<<<NSEG, 256, 0, stream>>>(bnd, sums, maxu, out);
}